// TOI_Pooling_6674379178726
// MI455X (gfx1250) — compile-verified
//
#include <hip/hip_runtime.h>
#include <stdint.h>

// ---- problem constants (from the reference) ----
#define B_  16
#define D_  256
#define L_  4096
#define N_  4096
#define TILE_W    64   // MAX_SPAN
#define ROW_PITCH 66   // 64 dwords + 2 dword TDM pad -> 8B-aligned, conflict-light LDS rows

typedef unsigned int v4u __attribute__((ext_vector_type(4)));
typedef int          v8i __attribute__((ext_vector_type(8)));
typedef int          v4i __attribute__((ext_vector_type(4)));
typedef float        v2f __attribute__((ext_vector_type(2)));
typedef float        v8f __attribute__((ext_vector_type(8)));

__global__ __launch_bounds__(256) void toi_pool_kernel(const float* __restrict__ features,
                                                       const int*   __restrict__ tois,
                                                       float*       __restrict__ out) {
  // 256 channels x (64 + 2 pad) dwords tile + 256-float staging row
  __shared__ float tile[D_ * ROW_PITCH];
  __shared__ float stage[D_];

  const int n   = blockIdx.x;
  const int b   = blockIdx.y;
  const int tid = threadIdx.x;
  const int q   = b * N_ + n;

  // toi is uniform per block; force into SGPRs for the TDM descriptor
  const int start = __builtin_amdgcn_readfirstlane(tois[2 * q]);
  const int span  = __builtin_amdgcn_readfirstlane(tois[2 * q + 1]) - start;

  const unsigned lds_base = (unsigned)(uintptr_t)(&tile[0]);

  // ---- Phase 1: wave 0 issues one TDM load of feat[b, 0:256, start:start+64] ----
  if (__builtin_amdgcn_readfirstlane(tid) == 0) {
    const uint64_t gaddr = (uint64_t)(uintptr_t)features +
                           4ull * ((uint64_t)b * (uint64_t)(D_ * L_) + (uint64_t)start);
    v4u g0;
    g0.x = 1u;                                           // count=1, user mode, no gather
    g0.y = lds_base;                                     // lds_addr (bytes)
    g0.z = (unsigned)(gaddr & 0xFFFFFFFFull);            // global_addr[31:0]
    g0.w = (unsigned)((gaddr >> 32) & 0x1FFFFFFull)      // global_addr[56:32]
         | (2u << 30);                                   // type=2 ("image")
    v8i g1;
    g1[0] = (2 << 16)        // data_size = 4 bytes
          | (1 << 20)        // pad_enable
          | (5 << 22)        // pad_interval: every 64 dwords
          | (1 << 25);       // pad_amount: 2 dwords  => row pitch 66 dwords
    g1[1] = (TILE_W << 16);  // [15:0] atomic_barrier_addr=0, [31:16] tensor_dim0=64
    g1[2] = (D_ << 16);      // [15:0] tensor_dim0 hi=0, [31:16] tensor_dim1=256
    g1[3] = (TILE_W << 16);  // [15:0] tensor_dim1 hi=0, [31:16] tile_dim0=64
    g1[4] = D_;              // [15:0] tile_dim1=256, [31:16] tile_dim2=0
    g1[5] = L_;              // tensor_dim0_stride[31:0] = 4096 elements
    g1[6] = 0;               // stride hi / tensor_dim1_stride lo
    g1[7] = 0;               // tensor_dim1_stride hi
    const v4i gz4 = {0, 0, 0, 0};
    const v8i gz8 = {0, 0, 0, 0, 0, 0, 0, 0};
    __builtin_amdgcn_tensor_load_to_lds(g0, g1, gz4, gz4, gz8, 0);
    __builtin_amdgcn_s_wait_tensorcnt(0);
  }
  __syncthreads();

  // ---- Phase 1b: zero the <=3 tail columns up to the next K=4 boundary so the
  //      WMMA loop needs no per-element masking (each thread owns its row).
  const int nsteps = (span + 3) >> 2;
  const int zend   = nsteps << 2;           // round span up to multiple of 4 (<=64)
  for (int c = span; c < zend; ++c) tile[tid * ROW_PITCH + c] = 0.0f;
  __syncthreads();

  // ---- Phase 2: span sums via V_WMMA_F32_16X16X4_F32 (B = ones), 2 groups/wave ----
  const int   wave = tid >> 5;
  const int   lane = tid & 31;
  const int   mrow = lane & 15;          // A-matrix row this lane carries
  const int   koff = (lane >> 4) << 1;   // lanes 0-15: K0,K1 ; lanes 16-31: K2,K3
  const float rinv = 1.0f / (float)span;
  const v2f   bones = {1.0f, 1.0f};      // all-ones B => every D column = row sum

  const int chbase = wave * 32;          // wave covers channels [chbase, chbase+32)
  const float* __restrict__ row0 = &tile[(chbase + mrow) * ROW_PITCH];
  const float* __restrict__ row1 = &tile[(chbase + 16 + mrow) * ROW_PITCH];

  v8f acc0 = {};
  v8f acc1 = {};
  for (int s = 0; s < nsteps; ++s) {
    const int kk = (s << 2) + koff;      // 8B-aligned: pitch even, kk even
    const v2f a0 = *(const v2f*)(row0 + kk);
    const v2f a1 = *(const v2f*)(row1 + kk);
    acc0 = __builtin_amdgcn_wmma_f32_16x16x4_f32(false, a0, false, bones,
                                                 (short)0, acc0, false, false);
    acc1 = __builtin_amdgcn_wmma_f32_16x16x4_f32(false, a1, false, bones,
                                                 (short)0, acc1, false, false);
  }

  // C layout: VGPR j -> M=j (lanes 0-15) / M=j+8 (lanes 16-31); lanes 0 & 16 stage avgs
  if ((lane & 15) == 0) {
    const int ob = chbase + ((lane >> 4) << 3);
#pragma unroll
    for (int j = 0; j < 8; ++j) {
      stage[ob + j]      = acc0[j] * rinv;
      stage[ob + 16 + j] = acc1[j] * rinv;
    }
  }
  __syncthreads();

  // ---- Phase 3: coalesced 768-float row write (head | avg | tail) ----
  const float head = tile[tid * ROW_PITCH];
  const float tail = tile[tid * ROW_PITCH + (span - 1)];
  const float avg  = stage[tid];
  float* __restrict__ orow = out + (size_t)q * (size_t)(3 * D_);
  orow[tid]          = head;
  orow[D_ + tid]     = avg;
  orow[2 * D_ + tid] = tail;

  // counts tail of d_out (tuple output, concatenated flat): cumsum of N per batch
  if (q == 0 && tid < B_) {
    out[(size_t)B_ * (size_t)N_ * (size_t)(3 * D_) + tid] = (float)((tid + 1) * N_);
  }
}

extern "C" void kernel_launch(void* const* d_in, const int* in_sizes, int n_in,
                              void* d_out, int out_size, void* d_ws, size_t ws_size,
                              hipStream_t stream) {
  (void)in_sizes; (void)n_in; (void)out_size; (void)d_ws; (void)ws_size;
  const float* features = (const float*)d_in[0];
  const int*   tois     = (const int*)d_in[1];
  float*       out      = (float*)d_out;
  dim3 grid(N_, B_);
  toi_pool_kernel<<<grid, 256, 0, stream>>>(features, tois, out);
}